// SparseInnerProductDecoder_79422535237675
// MI455X (gfx1250) — compile-verified
//
#include <hip/hip_runtime.h>

// ---------------------------------------------------------------------------
// SparseInnerProductDecoder for MI455X (gfx1250, wave32).
// out[e] = sigmoid( dot(z[row[e]], z[col[e]]) ),  z: [n,64] f32, E edges.
//
// Memory-bound kernel: z (25.6 MB) is L2-resident (192 MB L2), so we optimize
// for coalesced row fetches + cheap reduction. Each wave processes 16 edges:
//   - stage 16 row-rows + 16 col-rows (256B each) into padded LDS, coalesced
//   - 16x V_WMMA_F32_16X16X4_F32 accumulate a 16x16 Gram tile; its diagonal
//     is the 16 edge logits (full f32 precision, overcompute is free)
// ---------------------------------------------------------------------------

typedef float v2f __attribute__((ext_vector_type(2)));
typedef float v8f __attribute__((ext_vector_type(8)));

#define WAVES_PER_BLOCK 4
#define BLOCK_THREADS   (WAVES_PER_BLOCK * 32)
#define ROW_STRIDE      68   // 64 floats + 4 pad -> conflict-free LDS banking

__global__ __launch_bounds__(BLOCK_THREADS)
void sipd_wmma_kernel(const float* __restrict__ z,
                      const int*   __restrict__ eidx,   // [2, E] flat
                      float*       __restrict__ out,
                      int E, int nTiles)
{
    __shared__ float As[WAVES_PER_BLOCK][16 * ROW_STRIDE];
    __shared__ float Bs[WAVES_PER_BLOCK][16 * ROW_STRIDE];

    const int lane = threadIdx.x & 31;
    const int w    = threadIdx.x >> 5;
    const int sub  = lane & 15;   // M / N index within tile
    const int half = lane >> 4;   // K sub-slice selector

    const int* rowp = eidx;
    const int* colp = eidx + E;

    float* A = As[w];
    float* B = Bs[w];

    const int wavesTotal = gridDim.x * WAVES_PER_BLOCK;
    for (int tile = blockIdx.x * WAVES_PER_BLOCK + w; tile < nTiles; tile += wavesTotal) {
        const int e_base = tile * 16;

        // Each half-wave lane pair knows the indices for edge (e_base + sub).
        int e = e_base + sub;
        if (e >= E) e = E - 1;                 // tail clamp (E%16==0 here anyway)
        const int ridx = rowp[e];
        const int cidx = colp[e];

        // ---- Stage 32 rows (16 A + 16 B) of 64 floats into LDS, coalesced.
        // Per iteration each 16-lane group streams one full 256B row (float4/lane).
        #pragma unroll
        for (int it = 0; it < 8; ++it) {
            const int r  = it * 2 + half;                 // row slot 0..15
            const int ra = __shfl(ridx, r, 32);
            const int rb = __shfl(cidx, r, 32);
            const float4 va = *(const float4*)(z + (size_t)ra * 64 + sub * 4);
            const float4 vb = *(const float4*)(z + (size_t)rb * 64 + sub * 4);
            *(float4*)(A + r * ROW_STRIDE + sub * 4) = va;
            *(float4*)(B + r * ROW_STRIDE + sub * 4) = vb;
        }
        // Wave-internal produce->consume across lanes: wait for DS stores.
        asm volatile("s_wait_dscnt 0" ::: "memory");

        // ---- 16x16 Gram tile via chained V_WMMA_F32_16X16X4_F32 over K=64.
        // Fragment layout (A 16x4 and B 4x16 are addressed identically):
        //   lane L, element v  <-  mat[ idx[L%16] ][ kb + 2*(L/16) + v ]
        v8f acc = {0.f, 0.f, 0.f, 0.f, 0.f, 0.f, 0.f, 0.f};
        #pragma unroll
        for (int s = 0; s < 16; ++s) {
            const int kb = 4 * s;
            const v2f a = *(const v2f*)(A + sub * ROW_STRIDE + kb + 2 * half);
            const v2f b = *(const v2f*)(B + sub * ROW_STRIDE + kb + 2 * half);
            acc = __builtin_amdgcn_wmma_f32_16x16x4_f32(
                /*neg_a=*/false, a, /*neg_b=*/false, b,
                /*c_mod=*/(short)0, acc, /*reuse_a=*/false, /*reuse_b=*/false);
        }

        // ---- Extract diagonal D[m][m] (the 16 edge logits).
        // C/D layout: VGPR v holds M=v (lanes 0-15) and M=v+8 (lanes 16-31), N=lane%16.
        //   m = 0..7  -> acc[m]   at lane m
        //   m = 8..15 -> acc[m-8] at lane m+16  (lanes 24..31)
        float logit = 0.f;
        #pragma unroll
        for (int i = 0; i < 8; ++i) {
            const float ci = acc[i];
            logit = (lane == i || lane == i + 24) ? ci : logit;
        }

        const float sig = 1.0f / (1.0f + __expf(-logit));

        const bool writer = (lane < 8) | (lane >= 24);
        const int  m  = (lane < 8) ? lane : (lane - 16);
        const int  eo = e_base + m;
        if (writer && eo < E) out[eo] = sig;
    }
}

extern "C" void kernel_launch(void* const* d_in, const int* in_sizes, int n_in,
                              void* d_out, int out_size, void* d_ws, size_t ws_size,
                              hipStream_t stream) {
    const float* z    = (const float*)d_in[0];   // [n, 64] f32
    const int*   eidx = (const int*)d_in[1];     // [2, E] int32
    float*       out  = (float*)d_out;           // [E] f32

    const int E      = in_sizes[1] / 2;
    const int nTiles = (E + 15) / 16;
    const int blocks = (nTiles + WAVES_PER_BLOCK - 1) / WAVES_PER_BLOCK;

    sipd_wmma_kernel<<<blocks, BLOCK_THREADS, 0, stream>>>(z, eidx, out, E, nTiles);
}